// GaussianSliceRasterizer_79723182948527
// MI455X (gfx1250) — compile-verified
//
#include <hip/hip_runtime.h>
#include <math.h>

// ---------------- problem constants (from reference) ----------------
#define N_GAUSS   32768
#define NYPIX     128
#define NZPIX     128
#define NPIX      (NYPIX * NZPIX)       // 16384
#define NTILES    (NPIX / 16)           // 1024 pixel tiles of 16
#define SLICE_IDX 64

// px0 = CX - SX/2 + (SLICE_IDX+0.5)*(SX/NX) = -1 + 64.5/64
#define PX0       0.0078125f
#define LOG2E     1.4426950408889634f

typedef __attribute__((ext_vector_type(2))) float v2f;
typedef __attribute__((ext_vector_type(8))) float v8f;

// --------------------------------------------------------------------
// Per-Gaussian algebra: quaternion -> R, M = R*diag(s), cov = M M^T,
// analytic symmetric 3x3 inverse, then expand the quadratic form at
// fixed x = PX0 into pixel-space monomial coefficients:
//   quad(py,pz) = c0 + c1*py + c2*py^2 + c3*pz + c4*pz^2 + c5*py*pz
// Returned w[k] already folds in -0.5*log2(e); w[0] adds log2(opacity),
// so the field contribution is exactly exp2( w . monomials ).
// --------------------------------------------------------------------
__device__ __forceinline__ void gauss_coeffs(
    const float* __restrict__ means, const float* __restrict__ opac,
    const float* __restrict__ scales, const float* __restrict__ rots,
    int n, float w[6])
{
  float qw = rots[n * 4 + 0], qx = rots[n * 4 + 1];
  float qy = rots[n * 4 + 2], qz = rots[n * 4 + 3];
  float inr = rsqrtf(qw * qw + qx * qx + qy * qy + qz * qz);
  qw *= inr; qx *= inr; qy *= inr; qz *= inr;

  float r00 = 1.f - 2.f * (qy * qy + qz * qz);
  float r01 = 2.f * (qx * qy - qw * qz);
  float r02 = 2.f * (qx * qz + qw * qy);
  float r10 = 2.f * (qx * qy + qw * qz);
  float r11 = 1.f - 2.f * (qx * qx + qz * qz);
  float r12 = 2.f * (qy * qz - qw * qx);
  float r20 = 2.f * (qx * qz - qw * qy);
  float r21 = 2.f * (qy * qz + qw * qx);
  float r22 = 1.f - 2.f * (qx * qx + qy * qy);

  float s0 = scales[n * 3 + 0];            // SCALE_MODIFIER == 1.0
  float s1 = scales[n * 3 + 1];
  float s2 = scales[n * 3 + 2];
  float s0q = s0 * s0, s1q = s1 * s1, s2q = s2 * s2;

  // cov = R diag(s^2) R^T (symmetric)
  float cxx = r00 * r00 * s0q + r01 * r01 * s1q + r02 * r02 * s2q;
  float cxy = r00 * r10 * s0q + r01 * r11 * s1q + r02 * r12 * s2q;
  float cxz = r00 * r20 * s0q + r01 * r21 * s1q + r02 * r22 * s2q;
  float cyy = r10 * r10 * s0q + r11 * r11 * s1q + r12 * r12 * s2q;
  float cyz = r10 * r20 * s0q + r11 * r21 * s1q + r12 * r22 * s2q;
  float czz = r20 * r20 * s0q + r21 * r21 * s1q + r22 * r22 * s2q;

  // adjugate / det
  float m00 = cyy * czz - cyz * cyz;
  float m01 = cxz * cyz - cxy * czz;
  float m02 = cxy * cyz - cyy * cxz;
  float det = cxx * m00 + cxy * m01 + cxz * m02;
  float id  = 1.0f / det;
  float a = m00 * id;                      // inv[0][0]
  float b = m01 * id;                      // inv[0][1]
  float c = m02 * id;                      // inv[0][2]
  float d = (cxx * czz - cxz * cxz) * id;  // inv[1][1]
  float e = (cxy * cxz - cxx * cyz) * id;  // inv[1][2]
  float f = (cxx * cyy - cxy * cxy) * id;  // inv[2][2]

  float mx = means[n * 3 + 0], my = means[n * 3 + 1], mz = means[n * 3 + 2];
  float dx = PX0 - mx;

  float c0 = a * dx * dx + d * my * my + f * mz * mz
           - 2.f * b * dx * my - 2.f * c * dx * mz + 2.f * e * my * mz;
  float c1 = -2.f * d * my + 2.f * b * dx - 2.f * e * mz;   // * py
  float c2 = d;                                             // * py^2
  float c3 = -2.f * f * mz + 2.f * c * dx - 2.f * e * my;   // * pz
  float c4 = f;                                             // * pz^2
  float c5 = 2.f * e;                                       // * py*pz

  const float K = -0.5f * LOG2E;
  w[0] = K * c0 + log2f(opac[n]);
  w[1] = K * c1;  w[2] = K * c2;
  w[3] = K * c3;  w[4] = K * c4;  w[5] = K * c5;
}

// Pass 1: one thread per Gaussian -> coeff[n][8] (K padded to 8 with zeros)
__global__ void coeff_kernel(const float* __restrict__ means,
                             const float* __restrict__ opac,
                             const float* __restrict__ scales,
                             const float* __restrict__ rots,
                             float* __restrict__ coeff)
{
  int n = blockIdx.x * blockDim.x + threadIdx.x;
  if (n >= N_GAUSS) return;
  float w[6];
  gauss_coeffs(means, opac, scales, rots, n, w);
  float* o = coeff + (size_t)n * 8;
  o[0] = w[0]; o[1] = w[1]; o[2] = w[2];
  o[3] = w[3]; o[4] = w[4]; o[5] = w[5];
  o[6] = 0.f;  o[7] = 0.f;
}

// --------------------------------------------------------------------
// Pass 2: WMMA main loop.
// Each wave32 owns one 16-pixel tile (fixed y row, 16 consecutive z) and
// one Gaussian split. Per 16-Gaussian tile:
//   quad[16g x 16p] = A(16x8 coeffs) x B(8x16 monomials)   via 2x
//   V_WMMA_F32_16X16X4_F32, then 8x v_exp_f32 + add per lane.
// A layout (f32 16x4): v0 = K{0|2} by lane half, v1 = K{1|3}.
// B layout assumed symmetric: v0 = row K{0|2}, v1 = row K{1|3}, N = lane%16.
// D layout: lane holds N = lane%16, M = vgpr + 8*(lane/16)  -> fold the
// two lane halves with a shfl_xor(16) at the end.
// --------------------------------------------------------------------
__global__ void __launch_bounds__(256)
slice_wmma_kernel(const float* __restrict__ coeff,
                  float* __restrict__ outp,      // [nsplit][NPIX] or d_out
                  int gaussPerSplit)
{
  const int lane  = threadIdx.x & 31;
  const int wave  = blockIdx.x * (blockDim.x >> 5) + (threadIdx.x >> 5);
  const int tile  = wave & (NTILES - 1);
  const int split = wave >> 10;                  // NTILES == 1024
  const int half  = lane >> 4;                   // 0: lanes 0-15, 1: 16-31
  const int l16   = lane & 15;

  const int p0 = tile * 16;
  const int y  = p0 >> 7;
  const int z  = (p0 & 127) + l16;
  const float py = -1.0f + ((float)y + 0.5f) * (2.0f / 128.0f);
  const float pz = -1.0f + ((float)z + 0.5f) * (2.0f / 128.0f);

  // monomial rows of B for this lane's pixel column
  float mono[8] = {1.0f, py, py * py, pz, pz * pz, py * pz, 0.0f, 0.0f};
  v2f b0, b1;
  b0.x = mono[2 * half + 0]; b0.y = mono[2 * half + 1];
  b1.x = mono[2 * half + 4]; b1.y = mono[2 * half + 5];

  const float* base = coeff + (size_t)split * (size_t)gaussPerSplit * 8;
  const int ngtiles = gaussPerSplit >> 4;

  float accA = 0.0f, accB = 0.0f;
  for (int t = 0; t < ngtiles; ++t) {
    const float* gptr = base + ((size_t)t * 16 + (size_t)l16) * 8 + 2 * half;
    // prefetch next coefficient tile (global_prefetch_b8)
    __builtin_prefetch(gptr + 16 * 8, 0, 0);
    v2f a0 = *(const v2f*)(gptr);        // K = 2h, 2h+1
    v2f a1 = *(const v2f*)(gptr + 4);    // K = 4+2h, 4+2h+1

    v8f q = {};
    q = __builtin_amdgcn_wmma_f32_16x16x4_f32(false, a0, false, b0,
                                              (short)0, q, false, false);
    q = __builtin_amdgcn_wmma_f32_16x16x4_f32(false, a1, false, b1,
                                              (short)0, q, false, false);
    // 8 Gaussians' contributions for this lane's pixel
    accA += __builtin_amdgcn_exp2f(q[0]);
    accB += __builtin_amdgcn_exp2f(q[1]);
    accA += __builtin_amdgcn_exp2f(q[2]);
    accB += __builtin_amdgcn_exp2f(q[3]);
    accA += __builtin_amdgcn_exp2f(q[4]);
    accB += __builtin_amdgcn_exp2f(q[5]);
    accA += __builtin_amdgcn_exp2f(q[6]);
    accB += __builtin_amdgcn_exp2f(q[7]);
  }
  float acc = accA + accB;
  acc += __shfl_xor(acc, 16, 32);        // fold M=0..7 and M=8..15 halves
  if (half == 0)
    outp[(size_t)split * NPIX + p0 + l16] = acc;
}

// Pass 3: deterministic fixed-order reduction of the split partials
__global__ void reduce_kernel(const float* __restrict__ part,
                              float* __restrict__ outp, int nsplit)
{
  int p = blockIdx.x * blockDim.x + threadIdx.x;
  if (p >= NPIX) return;
  float s = 0.0f;
  for (int i = 0; i < nsplit; ++i) s += part[(size_t)i * NPIX + p];
  outp[p] = s;
}

// Safety fallback if workspace is too small for staging: one thread per
// pixel, recompute coefficients inline (slow, correct, no workspace).
__global__ void naive_kernel(const float* __restrict__ means,
                             const float* __restrict__ opac,
                             const float* __restrict__ scales,
                             const float* __restrict__ rots,
                             float* __restrict__ outp)
{
  int p = blockIdx.x * blockDim.x + threadIdx.x;
  if (p >= NPIX) return;
  int y = p >> 7, z = p & 127;
  float py = -1.0f + ((float)y + 0.5f) * (2.0f / 128.0f);
  float pz = -1.0f + ((float)z + 0.5f) * (2.0f / 128.0f);
  float acc = 0.0f;
  for (int n = 0; n < N_GAUSS; ++n) {
    float w[6];
    gauss_coeffs(means, opac, scales, rots, n, w);
    float q = w[0] + w[1] * py + w[2] * py * py + w[3] * pz +
              w[4] * pz * pz + w[5] * py * pz;
    acc += __builtin_amdgcn_exp2f(q);
  }
  outp[p] = acc;
}

extern "C" void kernel_launch(void* const* d_in, const int* in_sizes, int n_in,
                              void* d_out, int out_size, void* d_ws, size_t ws_size,
                              hipStream_t stream)
{
  const float* means  = (const float*)d_in[0];
  const float* opac   = (const float*)d_in[1];
  const float* scales = (const float*)d_in[2];
  const float* rots   = (const float*)d_in[3];
  float* out = (float*)d_out;

  const size_t coeffBytes = (size_t)N_GAUSS * 8 * sizeof(float);  // 1 MiB

  if (ws_size < coeffBytes) {
    // No room to stage coefficients: pure-VALU fallback.
    naive_kernel<<<(NPIX + 255) / 256, 256, 0, stream>>>(means, opac, scales,
                                                         rots, out);
    return;
  }

  float* coeffs = (float*)d_ws;
  coeff_kernel<<<(N_GAUSS + 255) / 256, 256, 0, stream>>>(means, opac, scales,
                                                          rots, coeffs);

  // Choose Gaussian-dimension split for occupancy (8192 waves at nsplit=8).
  int nsplit = 1;
  if (ws_size >= coeffBytes + (size_t)8 * NPIX * sizeof(float)) nsplit = 8;

  const int gaussPerSplit = N_GAUSS / nsplit;
  const int blocks = (NTILES * nsplit) / 8;  // 8 wave32 per 256-thread block

  if (nsplit == 1) {
    slice_wmma_kernel<<<blocks, 256, 0, stream>>>(coeffs, out, gaussPerSplit);
  } else {
    float* partials = (float*)((char*)d_ws + coeffBytes);
    slice_wmma_kernel<<<blocks, 256, 0, stream>>>(coeffs, partials,
                                                  gaussPerSplit);
    reduce_kernel<<<(NPIX + 255) / 256, 256, 0, stream>>>(partials, out,
                                                          nsplit);
  }
}